// NodeClassifier_19189913879016
// MI455X (gfx1250) — compile-verified
//
#include <hip/hip_runtime.h>

typedef __attribute__((ext_vector_type(2))) float v2f;
typedef __attribute__((ext_vector_type(8))) float v8f;

#define BS 256
#define BN_BLOCKS 256

// ---------------- degree / CSR build ----------------

__global__ void k_zero_cnt(int* cnt, int n) {
  int i = blockIdx.x * blockDim.x + threadIdx.x;
  if (i < n) cnt[i] = 0;
}

__global__ void k_count(const int* __restrict__ dst, int* __restrict__ cnt, int e) {
  int i = blockIdx.x * blockDim.x + threadIdx.x;
  if (i < e) atomicAdd(&cnt[dst[i]], 1);
}

__global__ void k_dinv(const int* __restrict__ cnt, float* __restrict__ dinv,
                       float* __restrict__ wself, int n) {
  int i = blockIdx.x * blockDim.x + threadIdx.x;
  if (i < n) {
    float d = 1.0f + (float)cnt[i];   // self loop included
    float di = rsqrtf(d);
    dinv[i] = di;
    wself[i] = di * di;
  }
}

// exclusive scan, 3 stages (N <= 131072 with BS=256, nblocks <= 512)
__global__ void k_scan1(const int* __restrict__ cnt, int* __restrict__ excl,
                        int* __restrict__ bsum, int n) {
  __shared__ int s[BS];
  int i = blockIdx.x * BS + threadIdx.x;
  int v = (i < n) ? cnt[i] : 0;
  s[threadIdx.x] = v;
  __syncthreads();
  for (int off = 1; off < BS; off <<= 1) {
    int t = (threadIdx.x >= off) ? s[threadIdx.x - off] : 0;
    __syncthreads();
    s[threadIdx.x] += t;
    __syncthreads();
  }
  if (i < n) excl[i] = s[threadIdx.x] - v;
  if (threadIdx.x == BS - 1) bsum[blockIdx.x] = s[BS - 1];
}

__global__ void k_scan2(const int* __restrict__ bsum, int* __restrict__ boff, int nb) {
  __shared__ int s[512];
  int v = ((int)threadIdx.x < nb) ? bsum[threadIdx.x] : 0;
  s[threadIdx.x] = v;
  __syncthreads();
  for (int off = 1; off < 512; off <<= 1) {
    int t = ((int)threadIdx.x >= off) ? s[threadIdx.x - off] : 0;
    __syncthreads();
    s[threadIdx.x] += t;
    __syncthreads();
  }
  if ((int)threadIdx.x < nb) boff[threadIdx.x] = s[threadIdx.x] - v;
}

__global__ void k_scan3(const int* __restrict__ excl, const int* __restrict__ boff,
                        int* __restrict__ rowst, int* __restrict__ cursor, int n) {
  int i = blockIdx.x * blockDim.x + threadIdx.x;
  if (i < n) {
    int r = excl[i] + boff[i >> 8];
    rowst[i] = r;
    cursor[i] = r;
  }
}

// CSR record: packed (src index, fp32 weight) -> one b64 load per edge in hot loops
__global__ void k_fill(const int* __restrict__ src, const int* __restrict__ dst,
                       const float* __restrict__ dinv, int* __restrict__ cursor,
                       int2* __restrict__ pairs, int e) {
  int i = blockIdx.x * blockDim.x + threadIdx.x;
  if (i < e) {
    int d = dst[i];
    int s = src[i];
    int pos = atomicAdd(&cursor[d], 1);
    int2 pr;
    pr.x = s;
    pr.y = __float_as_int(dinv[s] * dinv[d]);
    pairs[pos] = pr;
  }
}

// ---------------- propagation (wave per node, CSR gather) ----------------

__global__ void k_prop256(const float* __restrict__ hin, float* __restrict__ hout,
                          const int* __restrict__ rs, const int2* __restrict__ pairs,
                          const float* __restrict__ wself, int n, int etot) {
  int wid = (int)((blockIdx.x * blockDim.x + threadIdx.x) >> 5);
  int lane = threadIdx.x & 31;
  if (wid >= n) return;
  const float4* h4 = (const float4*)hin;
  float4 a0 = make_float4(0.f, 0.f, 0.f, 0.f);
  float4 a1 = make_float4(0.f, 0.f, 0.f, 0.f);
  int beg = rs[wid];
  int end = (wid + 1 < n) ? rs[wid + 1] : etot;
  for (int j = beg; j < end; ++j) {
    int2 pr = pairs[j];
    int s = pr.x;
    float w = __int_as_float(pr.y);
    float4 v0 = h4[(size_t)s * 64 + lane];
    float4 v1 = h4[(size_t)s * 64 + 32 + lane];
    a0.x = fmaf(w, v0.x, a0.x); a0.y = fmaf(w, v0.y, a0.y);
    a0.z = fmaf(w, v0.z, a0.z); a0.w = fmaf(w, v0.w, a0.w);
    a1.x = fmaf(w, v1.x, a1.x); a1.y = fmaf(w, v1.y, a1.y);
    a1.z = fmaf(w, v1.z, a1.z); a1.w = fmaf(w, v1.w, a1.w);
  }
  float ws = wself[wid];
  float4 u0 = h4[(size_t)wid * 64 + lane];
  float4 u1 = h4[(size_t)wid * 64 + 32 + lane];
  a0.x = fmaf(ws, u0.x, a0.x); a0.y = fmaf(ws, u0.y, a0.y);
  a0.z = fmaf(ws, u0.z, a0.z); a0.w = fmaf(ws, u0.w, a0.w);
  a1.x = fmaf(ws, u1.x, a1.x); a1.y = fmaf(ws, u1.y, a1.y);
  a1.z = fmaf(ws, u1.z, a1.z); a1.w = fmaf(ws, u1.w, a1.w);
  ((float4*)hout)[(size_t)wid * 64 + lane] = a0;
  ((float4*)hout)[(size_t)wid * 64 + 32 + lane] = a1;
}

__global__ void k_prop64(const float* __restrict__ hin, float* __restrict__ hout,
                         const int* __restrict__ rs, const int2* __restrict__ pairs,
                         const float* __restrict__ wself, int n, int etot) {
  int wid = (int)((blockIdx.x * blockDim.x + threadIdx.x) >> 5);
  int lane = threadIdx.x & 31;
  if (wid >= n) return;
  const float2* h2 = (const float2*)hin;
  float ax = 0.f, ay = 0.f;
  int beg = rs[wid];
  int end = (wid + 1 < n) ? rs[wid + 1] : etot;
  for (int j = beg; j < end; ++j) {
    int2 pr = pairs[j];
    int s = pr.x;
    float w = __int_as_float(pr.y);
    float2 v = h2[(size_t)s * 32 + lane];
    ax = fmaf(w, v.x, ax);
    ay = fmaf(w, v.y, ay);
  }
  float ws = wself[wid];
  float2 u = h2[(size_t)wid * 32 + lane];
  ax = fmaf(ws, u.x, ax);
  ay = fmaf(ws, u.y, ay);
  float2 r; r.x = ax; r.y = ay;
  ((float2*)hout)[(size_t)wid * 32 + lane] = r;
}

// ---------------- WMMA fp32 GEMM: C[n,NDIM] = A[n,KDIM] @ W[KDIM,NDIM] + bias ----------------
// One wave computes a 16-row x NDIM tile. V_WMMA_F32_16X16X4_F32, K stepped by 4.
// A layout: lane l holds A[row0 + l%16][k + 2*(l/16) + {0,1}]
// B layout: lane l holds W[k + 2*(l/16) + {0,1}][col0 + l%16]
// D layout: VGPR v, lane l -> row = row0 + (l/16)*8 + v, col = col0 + l%16

template <int KDIM, int NDIM>
__global__ void k_gemm_wmma(const float* __restrict__ A, const float* __restrict__ W,
                            const float* __restrict__ bias, float* __restrict__ C, int n) {
  int wid = (int)((blockIdx.x * blockDim.x + threadIdx.x) >> 5);
  int lane = threadIdx.x & 31;
  int row0 = wid * 16;
  if (row0 >= n) return;          // wave-uniform exit; EXEC all ones at the WMMA
  const int NT = NDIM / 16;
  int m = lane & 15;
  int half = lane >> 4;
  int row = row0 + m;
  if (row >= n) row = n - 1;      // clamp reads; boundary store handled below
  v8f zero = {};
  v8f acc[NT];
#pragma unroll
  for (int c = 0; c < NT; ++c) acc[c] = zero;

  for (int k = 0; k < KDIM; k += 4) {
    int ka = k + 2 * half;
    v2f a;
    a.x = A[(size_t)row * KDIM + ka];
    a.y = A[(size_t)row * KDIM + ka + 1];
#pragma unroll
    for (int c = 0; c < NT; ++c) {
      int col = c * 16 + m;
      v2f b;
      b.x = W[(size_t)ka * NDIM + col];
      b.y = W[(size_t)(ka + 1) * NDIM + col];
      acc[c] = __builtin_amdgcn_wmma_f32_16x16x4_f32(
          false, a, false, b, (short)0, acc[c], false, false);
    }
  }

  if (row0 + 16 <= n) {
    // full tile: unguarded store epilogue (all waves except possibly the last)
#pragma unroll
    for (int c = 0; c < NT; ++c) {
      int col = c * 16 + m;
      float bb = bias[col];
#pragma unroll
      for (int v = 0; v < 8; ++v) {
        int r = row0 + half * 8 + v;
        C[(size_t)r * NDIM + col] = acc[c][v] + bb;
      }
    }
  } else {
#pragma unroll
    for (int c = 0; c < NT; ++c) {
      int col = c * 16 + m;
      float bb = bias[col];
#pragma unroll
      for (int v = 0; v < 8; ++v) {
        int r = row0 + half * 8 + v;
        if (r < n) C[(size_t)r * NDIM + col] = acc[c][v] + bb;
      }
    }
  }
}

// ---------------- batchnorm (deterministic 2-stage) + SELU ----------------

__global__ void k_bnstats(const float* __restrict__ h, float* __restrict__ part, int n) {
  __shared__ float sh[256], sh2[256];
  int col = threadIdx.x & 63;
  int rsub = threadIdx.x >> 6;  // 0..3
  float s = 0.f, s2 = 0.f;
  for (int r = blockIdx.x * 4 + rsub; r < n; r += gridDim.x * 4) {
    float v = h[(size_t)r * 64 + col];
    s += v;
    s2 += v * v;
  }
  sh[threadIdx.x] = s;
  sh2[threadIdx.x] = s2;
  __syncthreads();
  if (rsub == 0) {
    s = sh[col] + sh[col + 64] + sh[col + 128] + sh[col + 192];
    s2 = sh2[col] + sh2[col + 64] + sh2[col + 128] + sh2[col + 192];
    part[blockIdx.x * 128 + col] = s;
    part[blockIdx.x * 128 + 64 + col] = s2;
  }
}

__global__ void k_bnfinal(const float* __restrict__ part, const float* __restrict__ gamma,
                          const float* __restrict__ beta, float* __restrict__ ss, int n) {
  int c = threadIdx.x;
  if (c >= 64) return;
  float s = 0.f, s2 = 0.f;
  for (int b = 0; b < BN_BLOCKS; ++b) {
    s += part[b * 128 + c];
    s2 += part[b * 128 + 64 + c];
  }
  float inv_n = 1.0f / (float)n;
  float mean = s * inv_n;
  float var = s2 * inv_n - mean * mean;
  float inv = rsqrtf(var + 1e-5f);
  float sc = gamma[c] * inv;
  ss[c] = sc;
  ss[64 + c] = beta[c] - mean * sc;
}

__global__ void k_bnselu(float* __restrict__ h, const float* __restrict__ ss, int total) {
  int i = blockIdx.x * blockDim.x + threadIdx.x;
  if (i >= total) return;
  int c = i & 63;
  float y = ss[c] * h[i] + ss[64 + c];
  const float alpha = 1.6732632423543772f;
  const float lam = 1.0507009873554805f;
  y = (y > 0.f) ? (lam * y) : (lam * alpha * (expf(y) - 1.f));
  h[i] = y;
}

// ---------------- log_softmax (wave per row of 32, in-place) ----------------

__global__ void k_logsoftmax(float* __restrict__ out, int n) {
  int wid = (int)((blockIdx.x * blockDim.x + threadIdx.x) >> 5);
  int lane = threadIdx.x & 31;
  if (wid >= n) return;
  float x = out[(size_t)wid * 32 + lane];
  float mx = x;
  for (int off = 16; off > 0; off >>= 1) mx = fmaxf(mx, __shfl_xor(mx, off, 32));
  float e = expf(x - mx);
  float s = e;
  for (int off = 16; off > 0; off >>= 1) s += __shfl_xor(s, off, 32);
  out[(size_t)wid * 32 + lane] = x - mx - logf(s);
}

// ---------------- launch ----------------

extern "C" void kernel_launch(void* const* d_in, const int* in_sizes, int n_in,
                              void* d_out, int out_size, void* d_ws, size_t ws_size,
                              hipStream_t stream) {
  const float* x     = (const float*)d_in[0];
  const int*   src   = (const int*)d_in[1];
  const int*   dst   = (const int*)d_in[2];
  const float* W1    = (const float*)d_in[3];
  const float* b1    = (const float*)d_in[4];
  const float* gamma = (const float*)d_in[5];
  const float* beta  = (const float*)d_in[6];
  const float* W2    = (const float*)d_in[7];
  const float* b2    = (const float*)d_in[8];
  float* out = (float*)d_out;

  const int N = in_sizes[0] / 256;
  const int E = in_sizes[1];

  char* base = (char*)d_ws;
  size_t off = 0;
  auto take = [&](size_t bytes) -> void* {
    void* p = base + off;
    off += (bytes + 255) & ~(size_t)255;
    return p;
  };

  int*   cnt    = (int*)take((size_t)N * 4);
  int*   excl   = (int*)take((size_t)N * 4);
  int*   bsum   = (int*)take(512 * 4);
  int*   boff   = (int*)take(512 * 4);
  int*   rowst  = (int*)take((size_t)N * 4);
  int*   cursor = (int*)take((size_t)N * 4);
  float* dinv   = (float*)take((size_t)N * 4);
  float* wself  = (float*)take((size_t)N * 4);
  int2*  pairs  = (int2*)take((size_t)E * 8);
  float* bnpart = (float*)take((size_t)BN_BLOCKS * 128 * 4);
  float* ss     = (float*)take(128 * 4);
  float* h1     = (float*)take((size_t)N * 256 * 4);
  float* h2     = (float*)take((size_t)N * 256 * 4);
  float* h64    = h1;                        // h1 region is free after gemm1 reads h2
  float* h64p   = h1 + (size_t)N * 64;

  const int gN  = (N + BS - 1) / BS;
  const int gE  = (E + BS - 1) / BS;
  const int nb1 = (N + BS - 1) / BS;          // scan blocks (<=512)
  const int gW  = (N + 7) / 8;                // wave-per-node kernels (8 waves/block)
  const int waves16 = (N + 15) / 16;
  const int gG  = (waves16 + 7) / 8;          // wave-per-16-rows GEMM kernels
  const int gEl = ((size_t)N * 64 + BS - 1) / BS;

  // CSR build
  k_zero_cnt<<<gN, BS, 0, stream>>>(cnt, N);
  k_count<<<gE, BS, 0, stream>>>(dst, cnt, E);
  k_dinv<<<gN, BS, 0, stream>>>(cnt, dinv, wself, N);
  k_scan1<<<nb1, BS, 0, stream>>>(cnt, excl, bsum, N);
  k_scan2<<<1, 512, 0, stream>>>(bsum, boff, nb1);
  k_scan3<<<gN, BS, 0, stream>>>(excl, boff, rowst, cursor, N);
  k_fill<<<gE, BS, 0, stream>>>(src, dst, dinv, cursor, pairs, E);

  // conv1: two propagation steps (256-dim, L2-resident gathers), then WMMA GEMM
  k_prop256<<<gW, BS, 0, stream>>>(x, h1, rowst, pairs, wself, N, E);
  k_prop256<<<gW, BS, 0, stream>>>(h1, h2, rowst, pairs, wself, N, E);
  k_gemm_wmma<256, 64><<<gG, BS, 0, stream>>>(h2, W1, b1, h64, N);

  // batchnorm + SELU
  k_bnstats<<<BN_BLOCKS, BS, 0, stream>>>(h64, bnpart, N);
  k_bnfinal<<<1, 64, 0, stream>>>(bnpart, gamma, beta, ss, N);
  k_bnselu<<<gEl, BS, 0, stream>>>(h64, ss, N * 64);

  // conv2: one propagation step (64-dim), WMMA GEMM straight into d_out
  k_prop64<<<gW, BS, 0, stream>>>(h64, h64p, rowst, pairs, wself, N, E);
  k_gemm_wmma<64, 32><<<gG, BS, 0, stream>>>(h64p, W2, b2, out, N);

  // row-wise log_softmax in place
  k_logsoftmax<<<gW, BS, 0, stream>>>(out, N);
}